// RandomShiftsAug_84696755077124
// MI455X (gfx1250) — compile-verified
//
#include <hip/hip_runtime.h>
#include <stdint.h>

// RandomShiftsAug: out[n,c,y,x] = in[n,c, clamp(y+dy), clamp(x+dx)]
// Pure bandwidth problem (~340 MB total -> ~15 us floor at 23.3 TB/s).
// Implemented entirely on the CDNA5 async global<->LDS DMA path (ASYNCcnt):
//   gather-in (async b32, shift applied in the global address) ->
//   s_wait_asynccnt / barrier ->
//   stream-out (async b128, aligned).
// Data never passes through VGPRs.

#define PADV 4
#define CCH  9
#define HH   96
#define WW   96
#define PLANE (HH * WW)                 // 9216 floats = 36 KB per plane
#define TPB  256
#define ELEMS_PER_THREAD (PLANE / TPB)  // 36
#define F4_PER_THREAD (PLANE / 4 / TPB) // 9

__global__ __launch_bounds__(TPB)
void RandomShiftsAug_kernel(const float* __restrict__ x,
                            const int* __restrict__ coords,
                            float* __restrict__ out)
{
    __shared__ float smem[PLANE];       // staged OUTPUT plane (post-gather)

    const int p   = blockIdx.x;         // plane index = n*C + c
    const int n   = p / CCH;
    const int tid = threadIdx.x;

    // Per-image uniform shift (scalar loads: address is block-uniform).
    const int dy = coords[2 * n + 0] - PADV;
    const int dx = coords[2 * n + 1] - PADV;

    const uint64_t src = (uint64_t)(uintptr_t)(x   + (size_t)p * PLANE);
    const uint64_t dst = (uint64_t)(uintptr_t)(out + (size_t)p * PLANE);

    // Raw LDS byte offset of smem: low 32 bits of the flat (generic) address
    // (AS3 offset occupies addr[31:0]; aperture bits live in addr[63:32]).
    const unsigned lbase = (unsigned)(uint64_t)(uintptr_t)(void*)smem;

    // ---- Stage 1: clamped gather, global -> LDS via async DMA (b32/lane) ----
    // Lanes cover consecutive output elements, so global addresses are
    // consecutive except at clamped edges -> ~128B contiguous per instruction.
    #pragma unroll
    for (int j = 0; j < ELEMS_PER_THREAD; ++j) {
        const int e  = tid + j * TPB;           // output element index in plane
        const int y  = e / WW;
        const int xx = e - y * WW;
        const int sy = min(max(y  + dy, 0), HH - 1);
        const int sx = min(max(xx + dx, 0), WW - 1);
        const unsigned goff = (unsigned)((sy * WW + sx) * 4); // src byte offset
        const unsigned loff = lbase + (unsigned)(e * 4);      // LDS byte addr
        asm volatile("global_load_async_to_lds_b32 %0, %1, %2"
                     :: "v"(loff), "v"(goff), "s"(src)
                     : "memory");
    }

    // Drain this wave's async loads, then make LDS visible across waves.
    asm volatile("s_wait_asynccnt 0" ::: "memory");
    __syncthreads();

    // ---- Stage 2: stream LDS -> global via async DMA (b128/lane) ----
    // Aligned 16B per lane, 512B per wave per instruction, fully coalesced.
    #pragma unroll
    for (int k = 0; k < F4_PER_THREAD; ++k) {
        const int q = tid + k * TPB;            // float4 index in plane
        const unsigned goff = (unsigned)(q * 16);
        const unsigned loff = lbase + (unsigned)(q * 16);
        asm volatile("global_store_async_from_lds_b128 %0, %1, %2"
                     :: "v"(goff), "v"(loff), "s"(dst)
                     : "memory");
    }
    // S_ENDPGM performs an implicit wait-idle (covers outstanding ASYNCcnt).
}

extern "C" void kernel_launch(void* const* d_in, const int* in_sizes, int n_in,
                              void* d_out, int out_size, void* d_ws, size_t ws_size,
                              hipStream_t stream) {
    const float* x      = (const float*)d_in[0];   // [N, C, H, W] fp32
    const int*   coords = (const int*)d_in[1];     // [N, 2] int32
    float*       out    = (float*)d_out;           // [N, C, H, W] fp32

    const int planes = in_sizes[0] / PLANE;        // N * C = 4608
    RandomShiftsAug_kernel<<<planes, TPB, 0, stream>>>(x, coords, out);
}